// RNNTLoss_21861383536653
// MI455X (gfx1250) — compile-verified
//
#include <hip/hip_runtime.h>
#include <stdint.h>

// Problem shape (fixed by the reference's setup_inputs).
#define BB 8
#define TT 256
#define UU 100
#define UO 101   // U+1
#define VV 128

// ---------------------------------------------------------------------------
// Kernel 1: fused log_softmax reduction + gather.
// One wave (32 lanes) per (b,t,u) row of 128 logits: each lane loads a float4
// (512B coalesced per wave), reduces max / sum(exp) across lanes with
// shuffles, then emits only the two values the DP needs:
//   lp_blank[b,t,u] = acts[b,t,u,0]           - lse
//   lp_emit [b,t,u] = acts[b,t,u,labels[b,u]] - lse   (u < U)
// Memory bound: reads 106 MB, writes ~1.6 MB.
// ---------------------------------------------------------------------------
__global__ __launch_bounds__(256) void rnnt_logprobs_kernel(
    const float* __restrict__ acts, const int* __restrict__ labels,
    float* __restrict__ lpB, float* __restrict__ lpE) {
  const int lane = threadIdx.x & 31;
  const int wid  = threadIdx.x >> 5;
  const int b    = blockIdx.y;
  const int idx  = blockIdx.x * 8 + wid;        // 0 .. T*U1-1 (exact tiling)
  const int t    = idx / UO;
  const int u    = idx - t * UO;

  const float4 v = reinterpret_cast<const float4*>(
      acts + (((size_t)b * TT + t) * UO + u) * VV)[lane];

  float m = fmaxf(fmaxf(v.x, v.y), fmaxf(v.z, v.w));
#pragma unroll
  for (int o = 16; o > 0; o >>= 1) m = fmaxf(m, __shfl_xor(m, o, 32));

  float s = expf(v.x - m) + expf(v.y - m) + expf(v.z - m) + expf(v.w - m);
#pragma unroll
  for (int o = 16; o > 0; o >>= 1) s += __shfl_xor(s, o, 32);

  const float lse = m + logf(s);
  const float a0  = __shfl(v.x, 0, 32);         // acts[...,0]

  float ev = 0.0f;
  if (u < UU) {
    const int lab  = labels[b * UU + u] & (VV - 1);   // wave-uniform
    const int comp = lab & 3;
    const float cand = comp == 0 ? v.x : comp == 1 ? v.y : comp == 2 ? v.z : v.w;
    ev = __shfl(cand, lab >> 2, 32);
  }

  if (lane == 0) {
    lpB[(size_t)b * (TT * UO) + idx] = a0 - lse;
    if (u < UU) lpE[(size_t)b * (TT * UU) + t * UU + u] = ev - lse;
  }
}

// ---------------------------------------------------------------------------
// Kernel 2: per-batch forward DP over anti-diagonals.
// One workgroup (128 threads = 4 wave32) per batch. The full per-batch
// blank (T*U1 = 103.4 KB) and emit (T*U = 102.4 KB) tables are staged into
// LDS with CDNA5 async-to-LDS B128 copies (ASYNCcnt), then all 355 diagonal
// steps run purely out of LDS (double-buffered diagonals).
// ---------------------------------------------------------------------------
__device__ __forceinline__ float log_add_exp(float a, float b) {
  const float mx = fmaxf(a, b);
  const float mn = fminf(a, b);
  return mx + log1pf(expf(mn - mx));
}

__global__ __launch_bounds__(128) void rnnt_alpha_kernel(
    const float* __restrict__ lpB, const float* __restrict__ lpE,
    const int* __restrict__ act_lens, const int* __restrict__ label_lens,
    float* __restrict__ wsLog) {
  extern __shared__ float smem[];
  float* sB  = smem;                       // T*U1 floats
  float* sE  = smem + TT * UO;             // T*U  floats
  float* dg0 = sE + TT * UU;               // diagonal buffer A (104)
  float* dg1 = dg0 + 104;                  // diagonal buffer B (104)

  const int tid = threadIdx.x;
  const int b   = blockIdx.x;

  const float* gB = lpB + (size_t)b * (TT * UO);
  const float* gE = lpE + (size_t)b * (TT * UU);

  // Generic pointer to LDS: low 32 bits are the workgroup-relative LDS byte
  // address expected by the async-load's VDST operand.
  const uint32_t ldsB = (uint32_t)(uintptr_t)sB;
  const uint32_t ldsE = (uint32_t)(uintptr_t)sE;

  // Async DMA of both tables into LDS (GVS form: vdst=LDS addr, vaddr=offset,
  // saddr=64-bit base). 6464 + 6400 B128 transfers spread over 128 threads.
  for (int i = tid; i < (TT * UO) / 4; i += 128) {
    const uint32_t la  = ldsB + (uint32_t)i * 16u;
    const int      off = i * 16;
    asm volatile("global_load_async_to_lds_b128 %0, %1, %2"
                 :: "v"(la), "v"(off), "s"(gB) : "memory");
  }
  for (int i = tid; i < (TT * UU) / 4; i += 128) {
    const uint32_t la  = ldsE + (uint32_t)i * 16u;
    const int      off = i * 16;
    asm volatile("global_load_async_to_lds_b128 %0, %1, %2"
                 :: "v"(la), "v"(off), "s"(gE) : "memory");
  }
  asm volatile("s_wait_asynccnt 0" ::: "memory");
  __syncthreads();

  const int tIdx = act_lens[b] - 1;
  const int uIdx = label_lens[b];

  float* prev = dg0;
  float* cur  = dg1;
  if (tid == 0) {
    prev[0] = 0.0f;                                  // alpha[0][0]
    if (tIdx == 0 && uIdx == 0) wsLog[b] = sB[0];    // degenerate terminal
  }
  __syncthreads();

  // Diagonal d holds cells (t = d-u, u); depends only on diagonal d-1.
  for (int d = 1; d <= TT - 1 + UU; ++d) {
    const int lo = (d - (TT - 1)) > 0 ? (d - (TT - 1)) : 0;
    const int hi = d < UU ? d : UU;
    const int u  = tid;
    if (u >= lo && u <= hi) {
      const int t = d - u;
      float val;
      if (t == 0) {                                   // row0 cumsum of emits
        val = prev[u - 1] + sE[u - 1];
      } else if (u == 0) {                            // blank-only column
        val = prev[0] + sB[(t - 1) * UO];
      } else {
        const float pa = prev[u]     + sB[(t - 1) * UO + u];
        const float pb = prev[u - 1] + sE[t * UU + (u - 1)];
        val = log_add_exp(pa, pb);
      }
      cur[u] = val;
      if (t == tIdx && u == uIdx)                     // terminal cell
        wsLog[b] = val + sB[t * UO + u];
    }
    __syncthreads();
    float* tmp = prev; prev = cur; cur = tmp;
  }
}

// ---------------------------------------------------------------------------
// Kernel 3: deterministic serial reduction of the B per-batch loglikes.
// ---------------------------------------------------------------------------
__global__ void rnnt_finalize_kernel(const float* __restrict__ wsLog,
                                     float* __restrict__ out) {
  if (threadIdx.x == 0 && blockIdx.x == 0) {
    float s = 0.0f;
    for (int b = 0; b < BB; ++b) s += wsLog[b];
    out[0] = -s / (float)BB;
  }
}

extern "C" void kernel_launch(void* const* d_in, const int* in_sizes, int n_in,
                              void* d_out, int out_size, void* d_ws, size_t ws_size,
                              hipStream_t stream) {
  const float* acts       = (const float*)d_in[0];
  const int*   labels     = (const int*)d_in[1];
  const int*   act_lens   = (const int*)d_in[2];
  const int*   label_lens = (const int*)d_in[3];
  float*       out        = (float*)d_out;

  // Workspace layout (16B-aligned segments):
  //   [0)                : lp_blank  B*T*U1 floats (827,392 B)
  //   [B*T*U1)           : lp_emit   B*T*U  floats (819,200 B)
  //   [B*T*U1 + B*T*U)   : per-batch loglike, B floats
  float* wsBlank = (float*)d_ws;
  float* wsEmit  = wsBlank + (size_t)BB * TT * UO;
  float* wsLog   = wsEmit  + (size_t)BB * TT * UU;

  // Phase 1: 8 waves/block, one wave per 128-logit row; exact tiling:
  // T*U1 = 25856 rows/batch = 3232 blocks of 8 waves.
  dim3 g1((TT * UO) / 8, BB);
  rnnt_logprobs_kernel<<<g1, 256, 0, stream>>>(acts, labels, wsBlank, wsEmit);

  // Phase 2: one workgroup per batch; dynamic LDS = blank + emit + 2 diagonals
  // = (25856 + 25600 + 208) * 4 = 206,656 B (< 320 KB/WGP on CDNA5).
  const size_t smemBytes = (size_t)(TT * UO + TT * UU + 2 * 104) * sizeof(float);
  rnnt_alpha_kernel<<<dim3(BB), 128, smemBytes, stream>>>(
      wsBlank, wsEmit, act_lens, label_lens, wsLog);

  // Phase 3: fixed-order reduction (deterministic across replays).
  rnnt_finalize_kernel<<<1, 32, 0, stream>>>(wsLog, out);
}